// GNNModelDGL_72421738545330
// MI455X (gfx1250) — compile-verified
//
#include <hip/hip_runtime.h>
#include <hip/hip_bf16.h>
#include <cmath>

#define NN 100000
#define NE 1600000

typedef float v2f __attribute__((ext_vector_type(2)));
typedef float v8f __attribute__((ext_vector_type(8)));

// ---------------------------------------------------------------------------
// float atomic-max via monotonic integer encodings (works with -inf init)
// ---------------------------------------------------------------------------
__device__ __forceinline__ void atomic_max_float(float* addr, float val) {
    if (val >= 0.0f) {
        atomicMax((int*)addr, __float_as_int(val));
    } else {
        atomicMin((unsigned int*)addr, __float_as_uint(val));
    }
}

__device__ __forceinline__ float lrelu02(float x) { return x > 0.0f ? x : 0.2f * x; }

// ---------------------------------------------------------------------------
// fill buffer with a constant (used for -inf init of segment-max buffers)
// ---------------------------------------------------------------------------
__global__ void fill_kernel(float* __restrict__ p, int n, float v) {
    int i = blockIdx.x * blockDim.x + threadIdx.x;
    if (i < n) p[i] = v;
}

// ---------------------------------------------------------------------------
// C[MxN] = A[MxK] @ B[KxN], row-major fp32 via V_WMMA_F32_16X16X4_F32.
// grid.x = M/16, blockDim.x = 32*(N/16); wave w computes tile (blockIdx.x, w).
// Requires M%16==0, N%16==0, K%4==0 (all exact here: M=100000=6250*16).
// EXEC is all-ones throughout (no divergence) as WMMA requires.
// ---------------------------------------------------------------------------
__global__ void wmma_gemm_f32(const float* __restrict__ A,
                              const float* __restrict__ B,
                              float* __restrict__ C,
                              int M, int N, int K) {
    const int lane  = threadIdx.x & 31;
    const int wave  = threadIdx.x >> 5;
    const int m0    = blockIdx.x * 16;
    const int n0    = wave * 16;
    const int mrow  = lane & 15;   // A: M on lanes
    const int khalf = lane >> 4;   // 0 -> K pair {0,1}, 1 -> K pair {2,3}
    const int ncol  = lane & 15;   // B/C/D: N on lanes

    v8f acc = {};
    const float* arow = A + (size_t)(m0 + mrow) * K;
    for (int k = 0; k < K; k += 4) {
        v2f a, b;
        const float* ap = arow + k + 2 * khalf;        // contiguous K pair
        a.x = ap[0];
        a.y = ap[1];
        const float* bp = B + (size_t)(k + 2 * khalf) * N + n0 + ncol;
        b.x = bp[0];
        b.y = bp[N];
        acc = __builtin_amdgcn_wmma_f32_16x16x4_f32(
            /*neg_a=*/false, a, /*neg_b=*/false, b,
            /*c_mod=*/(short)0, acc, /*reuse_a=*/false, /*reuse_b=*/false);
    }
    // D layout: VGPR r holds row (r + 8*khalf), col = lane&15
    float* crow = C + (size_t)(m0 + 8 * khalf) * N + n0 + ncol;
#pragma unroll
    for (int r = 0; r < 8; ++r) crow[(size_t)r * N] = acc[r];
}

// ---------------------------------------------------------------------------
// Layer-1 attention logits: el/er [N,8] from feat1 [N,128] (=[N,8,16])
// ---------------------------------------------------------------------------
__global__ void node_attn_l1(const float* __restrict__ feat,
                             const float* __restrict__ al,
                             const float* __restrict__ ar,
                             float* __restrict__ el, float* __restrict__ er) {
    int t = blockIdx.x * blockDim.x + threadIdx.x;
    if (t >= NN * 8) return;
    int n = t >> 3, h = t & 7;
    const float* f   = feat + (size_t)n * 128 + h * 16;
    const float* alh = al + h * 16;
    const float* arh = ar + h * 16;
    float sl = 0.0f, sr = 0.0f;
#pragma unroll
    for (int o = 0; o < 16; ++o) { float v = f[o]; sl += v * alh[o]; sr += v * arh[o]; }
    el[t] = sl; er[t] = sr;
}

// Layer-2 logits: el/er [N] from feat2 [N,64]
__global__ void node_attn_l2(const float* __restrict__ feat,
                             const float* __restrict__ al,
                             const float* __restrict__ ar,
                             float* __restrict__ el, float* __restrict__ er) {
    int n = blockIdx.x * blockDim.x + threadIdx.x;
    if (n >= NN) return;
    const float* f = feat + (size_t)n * 64;
    float sl = 0.0f, sr = 0.0f;
#pragma unroll 8
    for (int o = 0; o < 64; ++o) { float v = f[o]; sl += v * al[o]; sr += v * ar[o]; }
    el[n] = sl; er[n] = sr;
}

// ---------------------------------------------------------------------------
// Edge segment-max passes (atomicMax into emax[dst])
// ---------------------------------------------------------------------------
__global__ void edge_max_l1(const int* __restrict__ src, const int* __restrict__ dst,
                            const float* __restrict__ el, const float* __restrict__ er,
                            float* __restrict__ emax) {
    int t = blockIdx.x * blockDim.x + threadIdx.x;
    if (t >= NE * 8) return;
    int e = t >> 3, h = t & 7;
    int s = src[e], d = dst[e];
    float x = lrelu02(el[s * 8 + h] + er[d * 8 + h]);
    atomic_max_float(&emax[d * 8 + h], x);
}

__global__ void edge_max_l2(const int* __restrict__ src, const int* __restrict__ dst,
                            const float* __restrict__ el, const float* __restrict__ er,
                            float* __restrict__ emax) {
    int e = blockIdx.x * blockDim.x + threadIdx.x;
    if (e >= NE) return;
    int s = src[e], d = dst[e];
    float x = lrelu02(el[s] + er[d]);
    atomic_max_float(&emax[d], x);
}

// ---------------------------------------------------------------------------
// Edge accumulate passes: one wave per edge.
// L1: 128 dims -> 4 dims per lane, head = lane/4. Accumulate exp-weighted
// features into rstnum[dst] and exp-sum into denom[dst] (normalize later).
// ---------------------------------------------------------------------------
__global__ void edge_acc_l1(const int* __restrict__ src, const int* __restrict__ dst,
                            const float* __restrict__ el, const float* __restrict__ er,
                            const float* __restrict__ emax,
                            const float* __restrict__ feat,
                            float* __restrict__ denom, float* __restrict__ rstnum) {
    int lane = threadIdx.x & 31;
    int e = blockIdx.x * (blockDim.x >> 5) + (threadIdx.x >> 5);
    if (e >= NE) return;
    int s = src[e], d = dst[e];
    int h = lane >> 2;  // 4 lanes share one head
    float x  = lrelu02(el[s * 8 + h] + er[d * 8 + h]);
    float ex = __expf(x - emax[d * 8 + h]);
    if ((lane & 3) == 0) atomicAdd(&denom[d * 8 + h], ex);
    const float4 f = *(const float4*)(feat + (size_t)s * 128 + lane * 4);
    float* rn = rstnum + (size_t)d * 128 + lane * 4;
    atomicAdd(rn + 0, ex * f.x);
    atomicAdd(rn + 1, ex * f.y);
    atomicAdd(rn + 2, ex * f.z);
    atomicAdd(rn + 3, ex * f.w);
}

// L2: 64 dims -> 2 dims per lane, single head.
__global__ void edge_acc_l2(const int* __restrict__ src, const int* __restrict__ dst,
                            const float* __restrict__ el, const float* __restrict__ er,
                            const float* __restrict__ emax,
                            const float* __restrict__ feat,
                            float* __restrict__ denom, float* __restrict__ rstnum) {
    int lane = threadIdx.x & 31;
    int e = blockIdx.x * (blockDim.x >> 5) + (threadIdx.x >> 5);
    if (e >= NE) return;
    int s = src[e], d = dst[e];
    float x  = lrelu02(el[s] + er[d]);
    float ex = __expf(x - emax[d]);
    if (lane == 0) atomicAdd(&denom[d], ex);
    const float2 f = *(const float2*)(feat + (size_t)s * 64 + lane * 2);
    float* rn = rstnum + (size_t)d * 64 + lane * 2;
    atomicAdd(rn + 0, ex * f.x);
    atomicAdd(rn + 1, ex * f.y);
}

// ---------------------------------------------------------------------------
// Node finalize: L1 -> h = elu(rstnum/denom) (in-place safe, same index)
// ---------------------------------------------------------------------------
__global__ void finalize_l1(const float* __restrict__ rstnum,
                            const float* __restrict__ denom,
                            float* h) {
    int i = blockIdx.x * blockDim.x + threadIdx.x;
    if (i >= NN * 128) return;
    int n = i >> 7, dd = i & 127;
    float dn = denom[n * 8 + (dd >> 4)];
    float v = dn > 0.0f ? rstnum[i] / dn : 0.0f;       // empty segment -> 0
    h[i] = v > 0.0f ? v : __expf(v) - 1.0f;            // ELU (alpha=1)
}

// L2 -> out = rstnum/denom + residual
__global__ void finalize_l2(const float* __restrict__ rstnum,
                            const float* __restrict__ denom,
                            const float* __restrict__ res,
                            float* __restrict__ out) {
    int i = blockIdx.x * blockDim.x + threadIdx.x;
    if (i >= NN * 64) return;
    int n = i >> 6;
    float dn = denom[n];
    float v = dn > 0.0f ? rstnum[i] / dn : 0.0f;
    out[i] = v + res[i];
}

// ---------------------------------------------------------------------------
extern "C" void kernel_launch(void* const* d_in, const int* in_sizes, int n_in,
                              void* d_out, int out_size, void* d_ws, size_t ws_size,
                              hipStream_t stream) {
    const float* features = (const float*)d_in[0];
    const int*   src      = (const int*)d_in[1];
    const int*   dst      = (const int*)d_in[2];
    const float* W1       = (const float*)d_in[3];
    const float* al1      = (const float*)d_in[4];
    const float* ar1      = (const float*)d_in[5];
    const float* W2       = (const float*)d_in[6];
    const float* al2      = (const float*)d_in[7];
    const float* ar2      = (const float*)d_in[8];
    const float* Wres     = (const float*)d_in[9];
    float* out = (float*)d_out;

    // workspace layout (floats); ~142 MB total with aliasing
    float* w       = (float*)d_ws;
    float* feat1   = w;                              // N*128
    float* el1     = feat1   + (size_t)NN * 128;     // N*8
    float* er1     = el1     + (size_t)NN * 8;       // N*8
    float* emax1   = er1     + (size_t)NN * 8;       // N*8
    float* denom1  = emax1   + (size_t)NN * 8;       // N*8
    float* rstnum1 = denom1  + (size_t)NN * 8;       // N*128 (becomes h in place)
    float* rstnum2 = rstnum1 + (size_t)NN * 128;     // N*64
    float* el2     = rstnum2 + (size_t)NN * 64;      // N
    float* er2     = el2     + NN;                   // N
    float* emax2   = er2     + NN;                   // N
    float* denom2  = emax2   + NN;                   // N
    float* feat2   = feat1;                          // alias: feat1 dead after edge_acc_l1
    float* res     = feat1 + (size_t)NN * 64;        // alias: second half of feat1
    float* h       = rstnum1;                        // finalize_l1 is in-place

    // ---- init (graph-capture safe) ----
    hipMemsetAsync(denom1,  0, (size_t)NN * 8   * sizeof(float), stream);
    hipMemsetAsync(rstnum1, 0, (size_t)NN * 128 * sizeof(float), stream);
    hipMemsetAsync(denom2,  0, (size_t)NN       * sizeof(float), stream);
    hipMemsetAsync(rstnum2, 0, (size_t)NN * 64  * sizeof(float), stream);
    fill_kernel<<<(NN * 8 + 255) / 256, 256, 0, stream>>>(emax1, NN * 8, -INFINITY);
    fill_kernel<<<(NN + 255) / 256, 256, 0, stream>>>(emax2, NN, -INFINITY);

    // ---- layer 1: GATConv(128 -> 8x16), ELU ----
    wmma_gemm_f32<<<NN / 16, 32 * (128 / 16), 0, stream>>>(features, W1, feat1, NN, 128, 128);
    node_attn_l1<<<(NN * 8 + 255) / 256, 256, 0, stream>>>(feat1, al1, ar1, el1, er1);
    edge_max_l1<<<(NE * 8 + 255) / 256, 256, 0, stream>>>(src, dst, el1, er1, emax1);
    edge_acc_l1<<<NE / 8, 256, 0, stream>>>(src, dst, el1, er1, emax1, feat1, denom1, rstnum1);
    finalize_l1<<<(NN * 128 + 255) / 256, 256, 0, stream>>>(rstnum1, denom1, h);

    // ---- layer 2: GATConv(128 -> 64) + linear residual ----
    wmma_gemm_f32<<<NN / 16, 32 * (64 / 16), 0, stream>>>(h, W2,   feat2, NN, 64, 128);
    wmma_gemm_f32<<<NN / 16, 32 * (64 / 16), 0, stream>>>(h, Wres, res,   NN, 64, 128);
    node_attn_l2<<<(NN + 255) / 256, 256, 0, stream>>>(feat2, al2, ar2, el2, er2);
    edge_max_l2<<<(NE + 255) / 256, 256, 0, stream>>>(src, dst, el2, er2, emax2);
    edge_acc_l2<<<NE / 8, 256, 0, stream>>>(src, dst, el2, er2, emax2, feat2, denom2, rstnum2);
    finalize_l2<<<(NN * 64 + 255) / 256, 256, 0, stream>>>(rstnum2, denom2, res, out);
}